// ProdLayer_53197464928750
// MI455X (gfx1250) — compile-verified
//
#include <hip/hip_runtime.h>

// ProdLayer.forward (single group) for MI455X / gfx1250 (wave32).
//   element_mars[nids] = node_mars[cids].sum(axis=1)
// node_mars:    [NUM_NODES,   64] f32   (gather table, 256 MB -> mostly L2-resident)
// element_mars: [MAX_NUM_ELS, 64] f32   (output buffer)
// nids:         [n_group]      i32      (target rows, each written once)
// cids:         [n_group, 4]   i32      (child rows in node_mars)
//
// Pure bandwidth kernel: ~650 MB of traffic -> ~28 us floor at 23.3 TB/s.
// Design: 16 lanes x float4 per 256B row (wave32 = 2 rows/wave), 2 rows per
// thread for 8 in-flight global_load_b128 per lane, non-temporal stores so the
// streamed output does not evict the gather table from L2.

typedef __attribute__((ext_vector_type(4))) float v4f;
typedef __attribute__((ext_vector_type(4))) int   v4i;

static constexpr int BATCH   = 64;           // floats per row (reference constant)
static constexpr int VEC     = 4;            // float4
static constexpr int LPR     = BATCH / VEC;  // 16 lanes per row
static constexpr int ROWS_PT = 2;            // rows per thread

__global__ __launch_bounds__(256) void prod_fwd_kernel(
    const float* __restrict__ node_mars,
    const int*   __restrict__ nids,
    const int*   __restrict__ cids,
    float*       __restrict__ out,
    int n_group)
{
    const int tid  = blockIdx.x * blockDim.x + threadIdx.x;
    const int grp  = tid / LPR;        // row-group id (ROWS_PT rows each)
    const int q    = tid % LPR;        // float4 slot within the row
    const int b    = q * VEC;          // batch offset (16B aligned)
    const int row0 = grp * ROWS_PT;

    if (row0 >= n_group) return;

    // cids is streamed linearly by successive blocks; prefetch 512 rows (16 blocks) ahead.
    // Lowers to global_prefetch_b8 (speculative, no counter impact).
    if (q == 0) {
        __builtin_prefetch(cids + (size_t)(row0 + 2 * 256) * 4, 0, 1);
    }

#pragma unroll
    for (int rr = 0; rr < ROWS_PT; ++rr) {
        const int row = row0 + rr;
        if (row >= n_group) break;

        // 4 child indices for this product node: one aligned int4 load,
        // broadcast across the 16 lanes of the half-wave by the cache.
        const v4i c = ((const v4i*)cids)[row];

        // Four independent 256B-coalesced gathers (global_load_b128 x4 in
        // flight before the sum needs them). node_mars rows are 256B aligned,
        // +b*4 keeps 16B alignment. Regular temporal hint: we WANT these
        // cached in L2 (table is 256MB vs 192MB L2).
        const v4f a0 = *(const v4f*)(node_mars + (size_t)c.x * BATCH + b);
        const v4f a1 = *(const v4f*)(node_mars + (size_t)c.y * BATCH + b);
        const v4f a2 = *(const v4f*)(node_mars + (size_t)c.z * BATCH + b);
        const v4f a3 = *(const v4f*)(node_mars + (size_t)c.w * BATCH + b);

        const v4f s = (a0 + a1) + (a2 + a3);

        const int dst = nids[row];
        // Output is written once and never re-read by this dispatch:
        // non-temporal store keeps L2 free for the gather table.
        __builtin_nontemporal_store(s, (v4f*)(out + (size_t)dst * BATCH + b));
    }
}

// General-case fallback: seed the output with element_mars before scattering
// (only needed when the scatter does not provably cover every output row).
// NT on both sides: pure pass-through stream.
__global__ __launch_bounds__(256) void init_copy_kernel(
    const float* __restrict__ src,
    float*       __restrict__ dst,
    int n4)
{
    const int i = blockIdx.x * blockDim.x + threadIdx.x;
    if (i < n4) {
        v4f v = __builtin_nontemporal_load((const v4f*)src + i);
        __builtin_nontemporal_store(v, (v4f*)dst + i);
    }
}

extern "C" void kernel_launch(void* const* d_in, const int* in_sizes, int n_in,
                              void* d_out, int out_size, void* d_ws, size_t ws_size,
                              hipStream_t stream)
{
    const float* node_mars    = (const float*)d_in[0];
    const float* element_mars = (const float*)d_in[1];
    const int*   nids         = (const int*)  d_in[2];
    const int*   cids         = (const int*)  d_in[3];
    float*       out          = (float*)      d_out;

    const int n_group = in_sizes[2];

    // nids rows are each written once (reference contract). If the number of
    // scattered rows equals the number of output rows, the scatter covers the
    // whole buffer and the init copy would be 256MB of wasted traffic.
    const bool full_cover =
        ((long long)n_group * (long long)BATCH == (long long)out_size);

    if (!full_cover) {
        const int n4     = out_size / VEC;
        const int blocks = (n4 + 255) / 256;
        init_copy_kernel<<<blocks, 256, 0, stream>>>(element_mars, out, n4);
    }

    const int groups  = (n_group + ROWS_PT - 1) / ROWS_PT;
    const int threads = groups * LPR;
    const int blocks  = (threads + 255) / 256;
    prod_fwd_kernel<<<blocks, 256, 0, stream>>>(node_mars, nids, cids, out, n_group);
}